// MultiMipnet_14723147890783
// MI455X (gfx1250) — compile-verified
//
#include <hip/hip_runtime.h>
#include <math.h>

#define NUM_MODELS 64
#define IN_F 6
#define OUT_F 3
#define HID 128
#define BATCH 16384
#define MAXROWS (BATCH + NUM_MODELS * 15) /* 17344: per-model segments padded to 16 */
#define MAXTILES (MAXROWS / 16)           /* 1084 */
#define TILE_PAD 1088
#define LDSS 132                          /* LDS row stride (floats) for activations */

/* d_out layout (flat float32 concat of the reference tuple) */
#define OFF_MODELOUT 0
#define OFF_IDX      196608                       /* 16384*4*3 */
#define OFF_LOGITS   212992                       /* +16384 */
#define OFF_PROBS    1261568                      /* +16384*64 */

typedef __attribute__((ext_vector_type(2))) float v2f;
typedef __attribute__((ext_vector_type(8))) float v8f;
typedef __attribute__((ext_vector_type(4))) unsigned int v4u;
typedef __attribute__((ext_vector_type(8))) int v8i;
typedef __attribute__((ext_vector_type(4))) int v4i;

__global__ void k_init(int* __restrict__ hist, int* __restrict__ tile_model,
                       int* __restrict__ rowmap, float* __restrict__ gath) {
    int i = blockIdx.x * blockDim.x + threadIdx.x;
    int stride = gridDim.x * blockDim.x;
    for (int j = i; j < MAXROWS * 8; j += stride) gath[j] = 0.0f;
    for (int j = i; j < MAXROWS; j += stride) rowmap[j] = -1;
    for (int j = i; j < TILE_PAD; j += stride) tile_model[j] = -1;
    if (i < NUM_MODELS) hist[i] = 0;
}

__global__ void k_select(const float* __restrict__ in, int* __restrict__ idx32,
                         int* __restrict__ hist, float* __restrict__ out_idx) {
    int b = blockIdx.x * blockDim.x + threadIdx.x;
    if (b >= BATCH) return;
    float x = in[b * IN_F + 0];
    float z = in[b * IN_F + 2];
    float a = atan2f(z, x);
    /* mod(a + 2pi, 2pi) / 2pi * 64 -> floor */
    float u = a * 0.15915494309189535f + 1.0f; /* a/(2pi) + 1 */
    u = u - floorf(u);                          /* frac in [0,1) */
    int idx = (int)(u * (float)NUM_MODELS);
    if (idx < 0) idx = 0;
    if (idx > NUM_MODELS - 1) idx = NUM_MODELS - 1;
    idx32[b] = idx;
    atomicAdd(&hist[idx], 1);
    out_idx[b] = (float)idx;
}

__global__ void k_fill(float* __restrict__ logits, float* __restrict__ probs) {
    int i = blockIdx.x * blockDim.x + threadIdx.x;
    int stride = gridDim.x * blockDim.x;
    for (int j = i; j < BATCH * NUM_MODELS; j += stride) {
        logits[j] = 1.0f;
        probs[j]  = 1.0f / (float)NUM_MODELS;
    }
}

__global__ void k_prefix(const int* __restrict__ hist, int* __restrict__ offs,
                         int* __restrict__ cursor, int* __restrict__ tile_model) {
    if (threadIdx.x != 0 || blockIdx.x != 0) return;
    int run = 0, g = 0;
    for (int m = 0; m < NUM_MODELS; ++m) {
        offs[m] = run;
        cursor[m] = run;
        int nt = (hist[m] + 15) >> 4;
        for (int t = 0; t < nt; ++t) tile_model[g++] = m;
        run += nt << 4;
    }
}

__global__ void k_scatter(const float* __restrict__ in, const int* __restrict__ idx32,
                          int* __restrict__ cursor, int* __restrict__ rowmap,
                          float* __restrict__ gath) {
    int b = blockIdx.x * blockDim.x + threadIdx.x;
    if (b >= BATCH) return;
    int m = idx32[b];
    int pos = atomicAdd(&cursor[m], 1);
    rowmap[pos] = b;
#pragma unroll
    for (int j = 0; j < IN_F; ++j) gath[pos * 8 + j] = in[b * IN_F + j];
}

/* TDM: DMA a 16-row x S-col f32 tile of a row-major (stride HID) weight matrix
   into LDS at byte offset lds_off. pad_interval=32 DWORDs / pad_amount=1 DWORD
   swizzles element i to LDS dword (i + (i>>5)) -> conflict-free B-frag reads. */
__device__ __forceinline__ void tdm_load_wtile(const float* gsrc, unsigned lds_off, int S) {
    unsigned long long ga = (unsigned long long)(uintptr_t)gsrc;
    v4u g0;
    g0[0] = 1u;                                            /* count=1, user D# */
    g0[1] = lds_off;                                       /* lds_addr */
    g0[2] = (unsigned)ga;                                  /* global_addr[31:0] */
    g0[3] = ((unsigned)(ga >> 32) & 0x01FFFFFFu) | (2u << 30); /* addr[56:32] | type=2 */
    v8i g1;
    g1[0] = 0x01120000;          /* data_size=4B | pad_enable | pad_interval=32dw (amount=1dw) */
    g1[1] = (S & 0xFFFF) << 16;  /* tensor_dim0 = S   (atomic_barrier_addr=0) */
    g1[2] = 16 << 16;            /* tensor_dim1 = 16  (tensor_dim0 hi = 0) */
    g1[3] = S << 16;             /* tile_dim0 = S     (tensor_dim1 hi = 0) */
    g1[4] = 16;                  /* tile_dim1 = 16, tile_dim2 = 0 */
    g1[5] = HID;                 /* tensor_dim0_stride = 128 elements */
    g1[6] = 0;
    g1[7] = 0;
    v4i z4 = {0, 0, 0, 0};
    v8i z8 = {0, 0, 0, 0, 0, 0, 0, 0};
    __builtin_amdgcn_tensor_load_to_lds(g0, g1, z4, z4, z8, 0);
}

/* One wave32 per (16-row tile, LOD factor). Hidden layers via f32 WMMA 16x16x4,
   with weight tiles staged LDS-side by the Tensor Data Mover (double-buffered). */
__global__ __launch_bounds__(32) void k_mlp(
    const float* __restrict__ W0, const float* __restrict__ b0,
    const float* __restrict__ W1, const float* __restrict__ b1,
    const float* __restrict__ W2, const float* __restrict__ b2,
    const float* __restrict__ W3, const float* __restrict__ b3,
    const float* __restrict__ W4, const float* __restrict__ b4,
    const int* __restrict__ tile_model, const int* __restrict__ rowmap,
    const float* __restrict__ gath, float* __restrict__ out0) {
    __shared__ float Xa[16 * LDSS];
    __shared__ float Xb[16 * LDSS];
    __shared__ float Wt0[16 * LDSS]; /* padded weight tile: 16.5*S <= 2112 floats */
    __shared__ float Wt1[16 * LDSS];
    __shared__ float Xin[16 * 8];

    int bid = blockIdx.x;
    int f = bid & 3;       /* factor: S = 32*(f+1) */
    int t = bid >> 2;      /* tile index into padded gathered rows */
    int m = tile_model[t];
    if (m < 0) return;     /* uniform across wave */
    int S = 32 * (f + 1);
    int lane = threadIdx.x;

    /* LDS byte offsets of the two weight-tile buffers (runtime computed,
       no static-initializer of LDS addresses) */
    unsigned off0 = (unsigned)(uintptr_t)(void*)Wt0;
    unsigned off1 = (unsigned)(uintptr_t)(void*)Wt1;

    /* stage the 16x6 (padded-to-8) input rows */
    for (int i = lane; i < 16 * 8; i += 32) Xin[i] = gath[(t * 16) * 8 + i];
    __syncthreads();

    /* ---- layer 0: 6 -> S, scalar (K too small for WMMA) ---- */
    {
        const float* w0 = W0 + (size_t)m * HID * IN_F;
        const float* bb = b0 + (size_t)m * HID;
        for (int i = lane; i < 16 * S; i += 32) {
            int r = i & 15, c = i >> 4;
            float s = bb[c];
#pragma unroll
            for (int j = 0; j < IN_F; ++j) s += w0[c * IN_F + j] * Xin[r * 8 + j];
            Xa[r * LDSS + c] = fmaxf(s, 0.0f);
        }
    }
    __syncthreads();

    /* ---- hidden layers: S -> S via V_WMMA_F32_16X16X4_F32 ----
       A (16x4): lanes 0-15 rows; VGPR0 = K0/K2 halves, VGPR1 = K1/K3.
       B (4x16): same K striping, N = lane&15; read from TDM-staged LDS tile.
       C/D (16x16): VGPR v holds rows v (lanes 0-15) / v+8 (lanes 16-31). */
    const int col16 = lane & 15;
    const int ksel = (lane < 16) ? 0 : 2;
    auto hidden = [&](const float* __restrict__ W, const float* __restrict__ Bb,
                      const float* __restrict__ cur, float* __restrict__ nxt) {
        tdm_load_wtile(W, off0, S);                     /* prefetch first tile */
        for (int n = 0; n < S; n += 16) {
            int buf = (n >> 4) & 1;
            if (n + 16 < S) {
                tdm_load_wtile(W + (size_t)(n + 16) * HID, buf ? off0 : off1, S);
                __builtin_amdgcn_s_wait_tensorcnt(1);   /* in-order: tile n resident */
            } else {
                __builtin_amdgcn_s_wait_tensorcnt(0);
            }
            asm volatile("" ::: "memory");
            const float* Wt = buf ? Wt1 : Wt0;
            float bv = Bb[n + col16];
            v8f acc = {bv, bv, bv, bv, bv, bv, bv, bv};
            for (int k = 0; k < S; k += 4) {
                v2f afrag, bfrag;
                afrag.x = cur[col16 * LDSS + k + ksel];
                afrag.y = cur[col16 * LDSS + k + ksel + 1];
                int ib = col16 * S + k + ksel;          /* packed index in tile */
                const float* bp = Wt + ib + (ib >> 5);  /* TDM pad swizzle */
                bfrag.x = bp[0];
                bfrag.y = bp[1];
                acc = __builtin_amdgcn_wmma_f32_16x16x4_f32(
                    false, afrag, false, bfrag, (short)0, acc, false, false);
            }
#pragma unroll
            for (int v = 0; v < 8; ++v) {
                int row = v + ((lane < 16) ? 0 : 8);
                nxt[row * LDSS + n + col16] = fmaxf(acc[v], 0.0f);
            }
        }
    };

    hidden(W1 + (size_t)m * HID * HID, b1 + (size_t)m * HID, Xa, Xb);
    __syncthreads();
    hidden(W2 + (size_t)m * HID * HID, b2 + (size_t)m * HID, Xb, Xa);
    __syncthreads();
    hidden(W3 + (size_t)m * HID * HID, b3 + (size_t)m * HID, Xa, Xb);
    __syncthreads();

    /* ---- final layer: S -> 3, scalar, scatter through rowmap ---- */
    {
        const float* w4 = W4 + (size_t)m * OUT_F * HID;
        const float* bb = b4 + (size_t)m * OUT_F;
        for (int i = lane; i < 16 * OUT_F; i += 32) {
            int r = i / OUT_F, c = i % OUT_F;
            float s = bb[c];
            for (int k = 0; k < S; ++k) s += w4[c * HID + k] * Xb[r * LDSS + k];
            int b = rowmap[t * 16 + r];
            if (b >= 0) out0[(size_t)b * 12 + f * 3 + c] = s;
        }
    }
}

extern "C" void kernel_launch(void* const* d_in, const int* in_sizes, int n_in,
                              void* d_out, int out_size, void* d_ws, size_t ws_size,
                              hipStream_t stream) {
    (void)in_sizes; (void)n_in; (void)out_size; (void)ws_size;
    const float* in = (const float*)d_in[0];
    const float* W0 = (const float*)d_in[1];
    const float* b0 = (const float*)d_in[2];
    const float* W1 = (const float*)d_in[3];
    const float* b1 = (const float*)d_in[4];
    const float* W2 = (const float*)d_in[5];
    const float* b2 = (const float*)d_in[6];
    const float* W3 = (const float*)d_in[7];
    const float* b3 = (const float*)d_in[8];
    const float* W4 = (const float*)d_in[9];
    const float* b4 = (const float*)d_in[10];
    float* out = (float*)d_out;

    /* workspace partition */
    int* idx32      = (int*)d_ws;                 /* BATCH            */
    int* hist       = idx32 + BATCH;              /* NUM_MODELS       */
    int* offs       = hist + NUM_MODELS;          /* NUM_MODELS       */
    int* cursor     = offs + NUM_MODELS;          /* NUM_MODELS       */
    int* tile_model = cursor + NUM_MODELS;        /* TILE_PAD         */
    int* rowmap     = tile_model + TILE_PAD;      /* MAXROWS          */
    float* gath     = (float*)(rowmap + MAXROWS); /* MAXROWS*8 floats */

    k_init<<<544, 256, 0, stream>>>(hist, tile_model, rowmap, gath);
    k_select<<<(BATCH + 255) / 256, 256, 0, stream>>>(in, idx32, hist, out + OFF_IDX);
    k_fill<<<1024, 256, 0, stream>>>(out + OFF_LOGITS, out + OFF_PROBS);
    k_prefix<<<1, 1, 0, stream>>>(hist, offs, cursor, tile_model);
    k_scatter<<<(BATCH + 255) / 256, 256, 0, stream>>>(in, idx32, cursor, rowmap, gath);
    k_mlp<<<MAXTILES * 4, 32, 0, stream>>>(W0, b0, W1, b1, W2, b2, W3, b3, W4, b4,
                                           tile_model, rowmap, gath, out + OFF_MODELOUT);
}